// HypergraphPropagation_20109036880614
// MI455X (gfx1250) — compile-verified
//
#include <hip/hip_runtime.h>
#include <stdint.h>

#define V_NODES 100000
#define N_EDGES 20000
#define NNZ_CNT 1600000
#define DFEAT   256
#define EPSV    1e-6f

typedef unsigned int u32;
typedef u32 u32x4 __attribute__((ext_vector_type(4)));
typedef int i32x8 __attribute__((ext_vector_type(8)));
typedef int i32x4 __attribute__((ext_vector_type(4)));
typedef float f32x4 __attribute__((ext_vector_type(4)));

// ---------------------------------------------------------------- utilities

__global__ void zero_u32_kernel(u32* p, int n) {
    int i = blockIdx.x * blockDim.x + threadIdx.x;
    if (i < n) p[i] = 0u;
}

// Degrees (float sums of h_vals) + nnz histograms (for CSR build).
// COO streams are read once -> non-temporal loads keep them out of L2,
// preserving residency for X (102 MB) and HX (20 MB).
__global__ void degrees_kernel(const int* __restrict__ rows,
                               const int* __restrict__ cols,
                               const float* __restrict__ vals,
                               float* __restrict__ Dv, float* __restrict__ De,
                               u32* __restrict__ cntV, u32* __restrict__ cntE) {
    int i = blockIdx.x * blockDim.x + threadIdx.x;
    if (i >= NNZ_CNT) return;
    int r = __builtin_nontemporal_load(&rows[i]);
    int c = __builtin_nontemporal_load(&cols[i]);
    float v = __builtin_nontemporal_load(&vals[i]);
    atomicAdd(&Dv[r], v);
    atomicAdd(&De[c], v);
    atomicAdd(&cntV[r], 1u);
    atomicAdd(&cntE[c], 1u);
}

__global__ void scales_kernel(const float* __restrict__ Dv,
                              const float* __restrict__ De,
                              float* __restrict__ DvIS,
                              float* __restrict__ DeInv) {
    int i = blockIdx.x * blockDim.x + threadIdx.x;
    if (i < V_NODES) DvIS[i] = rsqrtf(Dv[i] + EPSV);
    if (i < N_EDGES) DeInv[i] = 1.0f / (De[i] + EPSV);
}

// Single-block exclusive scan (n up to ~100k; trivial runtime share).
__global__ __launch_bounds__(1024)
void scan_excl_kernel(const u32* __restrict__ counts,
                      u32* __restrict__ ptr, u32* __restrict__ cur, int n) {
    __shared__ u32 sh[1024];
    __shared__ u32 carry_sh;
    const int t = threadIdx.x;
    if (t == 0) carry_sh = 0u;
    __syncthreads();
    for (int base = 0; base < n; base += 1024) {
        u32 val = (base + t < n) ? counts[base + t] : 0u;
        sh[t] = val;
        __syncthreads();
        for (int off = 1; off < 1024; off <<= 1) {
            u32 add = (t >= off) ? sh[t - off] : 0u;
            __syncthreads();
            sh[t] += add;
            __syncthreads();
        }
        const u32 incl  = sh[t];
        const u32 carry = carry_sh;
        const u32 excl  = carry + incl - val;
        if (base + t < n) { ptr[base + t] = excl; cur[base + t] = excl; }
        __syncthreads();
        if (t == 1023) carry_sh = carry + sh[1023];
        __syncthreads();
    }
    if (t == 0) ptr[n] = carry_sh;
}

// COO -> CSR-by-edge and CSR-by-node (conflict-free gathers afterwards).
__global__ void build_csr_kernel(const int* __restrict__ rows,
                                 const int* __restrict__ cols,
                                 const float* __restrict__ vals,
                                 u32* __restrict__ curE, u32* __restrict__ curV,
                                 u32* __restrict__ erow, float* __restrict__ eval,
                                 u32* __restrict__ nedge, float* __restrict__ nval) {
    int i = blockIdx.x * blockDim.x + threadIdx.x;
    if (i >= NNZ_CNT) return;
    int r = __builtin_nontemporal_load(&rows[i]);
    int c = __builtin_nontemporal_load(&cols[i]);
    float v = __builtin_nontemporal_load(&vals[i]);
    u32 pe = atomicAdd(&curE[c], 1u);
    erow[pe] = (u32)r;
    eval[pe] = v;
    u32 pv = atomicAdd(&curV[r], 1u);
    nedge[pv] = (u32)c;
    nval[pv] = v;
}

// -------------------------------------------- TDM gather-mode row DMA to LDS
// Gathers up to 8 rows (32-bit indices) of a [V_NODES x 256] f32 tensor
// into LDS at lds_off, rows packed contiguously (1 KB each).
static __device__ inline void tdm_gather8(u32 lds_off, const float* base,
                                          int m, const int r[8]) {
    // D# group 0: count=1, gather_mode=1(bit31), 32-bit indices(bit30),
    //             lds_addr, 57-bit global_addr, type=2 (bits 127:126).
    uint64_t ga = (uint64_t)(uintptr_t)base;
    u32x4 g0;
    g0.x = 0xC0000001u;
    g0.y = lds_off;
    g0.z = (u32)ga;
    g0.w = ((u32)(ga >> 32) & 0x01FFFFFFu) | 0x80000000u;
    // D# group 1: workgroup_mask=0 (not in cluster), data_size=2 (4B),
    // tensor_dim0=256 (bits 79:48), tensor_dim1=V_NODES (bits 111:80),
    // tile_dim0=256 (bits 127:112), tile_dim1=m (#valid indices, bits 143:128),
    // tensor_dim0_stride=256 (bits 207:160).
    i32x8 g1;
    g1[0] = (int)(2u << 16);
    g1[1] = (int)((u32)DFEAT << 16);
    g1[2] = (int)(((u32)V_NODES & 0xFFFFu) << 16);
    g1[3] = (int)(((u32)V_NODES >> 16) | ((u32)DFEAT << 16));
    g1[4] = m;          // tile_dim1 = #indices, tile_dim2 = 0
    g1[5] = DFEAT;      // tensor_dim0_stride (low 32)
    g1[6] = 0;
    g1[7] = 0;
    // Gather row indices: D# groups 2/3 (two i32x4); the clang-23 6-arg
    // form adds a 256-bit operand — replicate the indices there.
    i32x4 g2 = { r[0], r[1], r[2], r[3] };
    i32x4 g3 = { r[4], r[5], r[6], r[7] };
    i32x8 g23 = { r[0], r[1], r[2], r[3], r[4], r[5], r[6], r[7] };
    __builtin_amdgcn_tensor_load_to_lds(g0, g1, g2, g3, g23, 0);
}

static __device__ inline void issue_chunk(u32 lds_off, const float* X,
                                          const u32* __restrict__ erow,
                                          u32 b, u32 end) {
    int m = (int)(end - b);
    if (m > 8) m = 8;
    int r[8];
#pragma unroll
    for (int k = 0; k < 8; ++k)
        r[k] = (k < m) ? __builtin_amdgcn_readfirstlane((int)erow[b + k]) : 0;
    tdm_gather8(lds_off, X, m, r);
}

// ------------------------------------------------------- pass 1: HX = De^-1 H^T Xs
// One wave32 per hyperedge. Member-node rows (avg 80/edge) are DMA-gathered
// 8 at a time into double-buffered LDS via the TDM gather mode, overlapped
// with the accumulation of the previous chunk (s_wait_tensorcnt 1).
__global__ __launch_bounds__(32)
void pass1_edges_kernel(const u32* __restrict__ ptrE,
                        const u32* __restrict__ erow,
                        const float* __restrict__ eval,
                        const float* __restrict__ X,
                        const float* __restrict__ DvIS,
                        const float* __restrict__ DeInv,
                        float* __restrict__ HX) {
    __shared__ float buf[2][8 * DFEAT];          // 2 x 8 KB double buffer
    const int e    = blockIdx.x;
    const int lane = threadIdx.x;
    const u32 beg  = ptrE[e];
    const u32 end  = ptrE[e + 1];
    const int nch  = ((int)(end - beg) + 7) >> 3;
    const u32 lds0 = (u32)(uintptr_t)(&buf[0][0]);

    float a0 = 0.f, a1 = 0.f, a2 = 0.f, a3 = 0.f;
    float b0 = 0.f, b1 = 0.f, b2 = 0.f, b3 = 0.f;

    if (nch > 0) issue_chunk(lds0, X, erow, beg, end);
    for (int c = 0; c < nch; ++c) {
        const bool more = (c + 1) < nch;
        if (more) {
            issue_chunk(lds0 + (u32)(((c + 1) & 1) * 8 * DFEAT * 4),
                        X, erow, beg + (u32)((c + 1) << 3), end);
            __builtin_amdgcn_s_wait_tensorcnt(1);   // chunk c has landed
        } else {
            __builtin_amdgcn_s_wait_tensorcnt(0);
        }
        const float* lb = &buf[c & 1][0];
        const u32 b = beg + (u32)(c << 3);
        int m = (int)(end - b);
        if (m > 8) m = 8;
        for (int k = 0; k < m; ++k) {
            const int   row = __builtin_amdgcn_readfirstlane((int)erow[b + k]);
            const float s   = eval[b + k] * DvIS[row];      // fold Dv^-1/2 here
            const f32x4 xa = *(const f32x4*)&lb[k * DFEAT + lane * 4];
            const f32x4 xb = *(const f32x4*)&lb[k * DFEAT + 128 + lane * 4];
            a0 += s * xa.x; a1 += s * xa.y; a2 += s * xa.z; a3 += s * xa.w;
            b0 += s * xb.x; b1 += s * xb.y; b2 += s * xb.z; b3 += s * xb.w;
        }
    }
    const float di = DeInv[e];
    f32x4 o0 = { a0 * di, a1 * di, a2 * di, a3 * di };
    f32x4 o1 = { b0 * di, b1 * di, b2 * di, b3 * di };
    float* out = HX + (size_t)e * DFEAT;         // re-read by pass 2: keep RT
    *(f32x4*)&out[lane * 4]       = o0;
    *(f32x4*)&out[128 + lane * 4] = o1;
}

// ------------------------------------------------- pass 2: out = Dv^-1/2 H HX
// One wave32 per node (avg 16 member edges). HX is 20 MB -> L2 resident.
// Final output (102 MB) is written once and never re-read: non-temporal
// stores avoid evicting the X/HX working set from L2.
__global__ __launch_bounds__(32)
void pass2_nodes_kernel(const u32* __restrict__ ptrV,
                        const u32* __restrict__ nedge,
                        const float* __restrict__ nval,
                        const float* __restrict__ HX,
                        const float* __restrict__ DvIS,
                        float* __restrict__ out) {
    const int v    = blockIdx.x;
    const int lane = threadIdx.x;
    const u32 beg  = ptrV[v];
    const u32 end  = ptrV[v + 1];

    float a0 = 0.f, a1 = 0.f, a2 = 0.f, a3 = 0.f;
    float b0 = 0.f, b1 = 0.f, b2 = 0.f, b3 = 0.f;

    for (u32 j = beg; j < end; ++j) {
        const int   e = __builtin_amdgcn_readfirstlane((int)nedge[j]);
        const float w = nval[j];
        const float* row = HX + (size_t)e * DFEAT;
        const f32x4 xa = *(const f32x4*)&row[lane * 4];
        const f32x4 xb = *(const f32x4*)&row[128 + lane * 4];
        a0 += w * xa.x; a1 += w * xa.y; a2 += w * xa.z; a3 += w * xa.w;
        b0 += w * xb.x; b1 += w * xb.y; b2 += w * xb.z; b3 += w * xb.w;
    }
    const float sc = DvIS[v];
    f32x4 o0 = { a0 * sc, a1 * sc, a2 * sc, a3 * sc };
    f32x4 o1 = { b0 * sc, b1 * sc, b2 * sc, b3 * sc };
    f32x4* op0 = (f32x4*)(out + (size_t)v * DFEAT + lane * 4);
    f32x4* op1 = (f32x4*)(out + (size_t)v * DFEAT + 128 + lane * 4);
    __builtin_nontemporal_store(o0, op0);
    __builtin_nontemporal_store(o1, op1);
}

// ---------------------------------------------------------------- launcher

extern "C" void kernel_launch(void* const* d_in, const int* in_sizes, int n_in,
                              void* d_out, int out_size, void* d_ws, size_t ws_size,
                              hipStream_t stream) {
    (void)in_sizes; (void)n_in; (void)out_size; (void)ws_size;
    const float* X    = (const float*)d_in[0];
    const int*   rows = (const int*)d_in[1];
    const int*   cols = (const int*)d_in[2];
    const float* vals = (const float*)d_in[3];
    float* out = (float*)d_out;

    // Workspace carve-up. Zero-initialized block (Dv, De, cntE, cntV) first.
    char* w = (char*)d_ws;
    float* Dv    = (float*)w; w += (size_t)V_NODES * 4;
    float* De    = (float*)w; w += (size_t)N_EDGES * 4;
    u32*   cntE  = (u32*)w;   w += (size_t)N_EDGES * 4;
    u32*   cntV  = (u32*)w;   w += (size_t)V_NODES * 4;
    float* DvIS  = (float*)w; w += (size_t)V_NODES * 4;
    float* DeInv = (float*)w; w += (size_t)N_EDGES * 4;
    u32*   ptrE  = (u32*)w;   w += (size_t)(N_EDGES + 1) * 4;
    u32*   ptrV  = (u32*)w;   w += (size_t)(V_NODES + 1) * 4;
    u32*   curE  = (u32*)w;   w += (size_t)N_EDGES * 4;
    u32*   curV  = (u32*)w;   w += (size_t)V_NODES * 4;
    u32*   erow  = (u32*)w;   w += (size_t)NNZ_CNT * 4;
    float* eval  = (float*)w; w += (size_t)NNZ_CNT * 4;
    u32*   nedge = (u32*)w;   w += (size_t)NNZ_CNT * 4;
    float* nval  = (float*)w; w += (size_t)NNZ_CNT * 4;
    float* HX    = (float*)w; w += (size_t)N_EDGES * DFEAT * 4;

    const int zwords = 2 * (V_NODES + N_EDGES);
    zero_u32_kernel<<<(zwords + 255) / 256, 256, 0, stream>>>((u32*)d_ws, zwords);

    degrees_kernel<<<(NNZ_CNT + 255) / 256, 256, 0, stream>>>(
        rows, cols, vals, Dv, De, cntV, cntE);

    scales_kernel<<<(V_NODES + 255) / 256, 256, 0, stream>>>(Dv, De, DvIS, DeInv);

    scan_excl_kernel<<<1, 1024, 0, stream>>>(cntE, ptrE, curE, N_EDGES);
    scan_excl_kernel<<<1, 1024, 0, stream>>>(cntV, ptrV, curV, V_NODES);

    build_csr_kernel<<<(NNZ_CNT + 255) / 256, 256, 0, stream>>>(
        rows, cols, vals, curE, curV, erow, eval, nedge, nval);

    pass1_edges_kernel<<<N_EDGES, 32, 0, stream>>>(
        ptrE, erow, eval, X, DvIS, DeInv, HX);

    pass2_nodes_kernel<<<V_NODES, 32, 0, stream>>>(
        ptrV, nedge, nval, HX, DvIS, out);
}